// NnHalfKA_29566554865846
// MI455X (gfx1250) — compile-verified
//
#include <hip/hip_runtime.h>
#include <math.h>

// ---------------------------------------------------------------------------
// NNUE (HalfKA) forward for MI455X / gfx1250 (wave32).
//
//  1) Transpose ft_w (1024x49152) -> ftT (49152x1024) in workspace: gathers
//     become contiguous 4KB rows; table is L2-resident (192MB vs 192MB L2).
//  2) Virtual-feature term as dense f32 GEMM with V_WMMA_F32_16X16X4_F32:
//     - fft_w packed K-pair-interleaved (fftP[k/2][n][2]) so each lane's
//       B-fragment is one contiguous b64 load.
//     - both perspectives' vboards staged in LDS (dynamic, ~98.8KB) so each
//       B fragment feeds two WMMAs (stm + nstm) -> halves B-side L2 traffic.
//  3) Per-row sparse gather + bias + clamp + out_w dot + wave32 reduction.
//  4) Bucket-indexed sigmoid.
//
// Workspace requirement: ~238 MB (ftT 192MB + fftP 3MB + fftC 32MB + l1).
// ---------------------------------------------------------------------------

#define NN_B 4096
#define NN_NNZ 32
#define NN_FTIN 49152
#define NN_VIRT 768
#define NN_FTOUT 1024
#define VB_STRIDE 772 // 768 + 4 pad: avoids LDS bank conflicts, keeps 16B align

typedef __attribute__((ext_vector_type(2))) float v2f;
typedef __attribute__((ext_vector_type(8))) float v8f;

// ---------------- tiled transpose: in[R][C] -> out[C][R] -------------------
__global__ __launch_bounds__(256) void nn_transpose(const float* __restrict__ in,
                                                    float* __restrict__ out,
                                                    int R, int C) {
  __shared__ float t[32][33];
  const int bx = blockIdx.x * 32; // column tile
  const int by = blockIdx.y * 32; // row tile
  const int tx = threadIdx.x;     // 0..31
  for (int i = threadIdx.y; i < 32; i += 8) {
    int y = by + i, x = bx + tx;
    t[i][tx] = (y < R && x < C) ? in[(size_t)y * C + x] : 0.0f;
  }
  __syncthreads();
  for (int i = threadIdx.y; i < 32; i += 8) {
    int yo = bx + i, xo = by + tx;
    if (yo < C && xo < R) out[(size_t)yo * R + xo] = t[tx][i];
  }
}

// ------- pack fft_w (FT_OUT x VIRT) into K-pair-interleaved transpose ------
// fftP[(k>>1)*FT_OUT*2 + n*2 + (k&1)] = fft_w[n][k]  => lane B-frag = 1x b64
__global__ __launch_bounds__(256) void nn_pack_fft(const float* __restrict__ fft_w,
                                                   float* __restrict__ fftP) {
  const int i = blockIdx.x * 256 + threadIdx.x; // coalesced read over n*768+k
  if (i < NN_VIRT * NN_FTOUT) {
    const int n = i / NN_VIRT;
    const int k = i % NN_VIRT;
    fftP[((size_t)(k >> 1) * NN_FTOUT + n) * 2 + (k & 1)] = fft_w[i];
  }
}

// -------- virtual-feature dense GEMM via f32 WMMA (16-row batch tiles) -----
// Both perspectives share B fragments: one b64 B load feeds two WMMAs.
// fftC layout: [persp][B][FT_OUT] (f32), fully written (no pre-zero needed).
__global__ __launch_bounds__(256) void nn_fft_wmma(const int* __restrict__ stm_idx,
                                                   const int* __restrict__ nstm_idx,
                                                   const float* __restrict__ values,
                                                   const float* __restrict__ fftP,
                                                   float* __restrict__ fftC) {
  extern __shared__ float vb[]; // [2][16][VB_STRIDE] = 98,816 bytes (dynamic)
  const int row0 = blockIdx.x * 16;
  const int tid  = threadIdx.x;
  const int wave = tid >> 5;
  const int lane = tid & 31;
  const int half = lane >> 4;   // 0: lanes 0-15, 1: lanes 16-31
  const int m    = lane & 15;   // A-matrix row / B,C column within tile
  // per ISA f32 16x4 A layout: lane half selects K pair (K = 2*half, 2*half+1)

  // zero + scatter both perspectives' vboard tiles (one thread per row/persp)
  for (int i = tid; i < 2 * 16 * VB_STRIDE; i += 256) vb[i] = 0.0f;
  __syncthreads();
  if (tid < 32) {
    const int persp = tid >> 4;
    const int r     = tid & 15;
    const int* idx  = persp ? nstm_idx : stm_idx;
    const int base  = (row0 + r) * NN_NNZ;
    float* row = vb + (persp * 16 + r) * VB_STRIDE;
    for (int k = 0; k < NN_NNZ; ++k) {
      int f = idx[(base + k) * 2 + 1];
      row[f % NN_VIRT] += values[base + k];
    }
  }
  __syncthreads();

  const float* vb0 = vb + m * VB_STRIDE + 2 * half;       // stm A row
  const float* vb1 = vb0 + 16 * VB_STRIDE;                // nstm A row
  float* o0 = fftC + (size_t)row0 * NN_FTOUT;
  float* o1 = fftC + ((size_t)NN_B + row0) * NN_FTOUT;
  const int rsel = half * 8;

  // each wave covers 128 output columns = 8 tiles of 16
  for (int ct = 0; ct < 8; ++ct) {
    const int cbase = wave * 128 + ct * 16;
    // pair index for this lane: k2 = (kb>>1) + half, element n = cbase+m
    const v2f* bptr = (const v2f*)fftP + ((size_t)half * NN_FTOUT + cbase + m);
    v8f c0 = {}, c1 = {};
    for (int kb = 0; kb < NN_VIRT; kb += 4) {
      v2f b = bptr[(size_t)(kb >> 1) * NN_FTOUT]; // contiguous b64 per lane
      v2f a0, a1;
      a0.x = vb0[kb]; a0.y = vb0[kb + 1];
      a1.x = vb1[kb]; a1.y = vb1[kb + 1];
      c0 = __builtin_amdgcn_wmma_f32_16x16x4_f32(false, a0, false, b, (short)0,
                                                 c0, false, false);
      c1 = __builtin_amdgcn_wmma_f32_16x16x4_f32(false, a1, false, b, (short)0,
                                                 c1, false, false);
    }
    // C/D layout: VGPR i -> row (i + 8*half), lanes 0-15/16-31 -> col m
#pragma unroll
    for (int i = 0; i < 8; ++i) {
      o0[(size_t)(rsel + i) * NN_FTOUT + cbase + m] = c0[i];
      o1[(size_t)(rsel + i) * NN_FTOUT + cbase + m] = c1[i];
    }
  }
}

// -------- per-row sparse FT gather + bias + clamp + out_w dot --------------
__global__ __launch_bounds__(256) void nn_ft_final(const int* __restrict__ stm_idx,
                                                   const int* __restrict__ nstm_idx,
                                                   const float* __restrict__ values,
                                                   const float* __restrict__ ftT,
                                                   const float* __restrict__ ft_b,
                                                   const float* __restrict__ fft_b,
                                                   const float* __restrict__ fftC,
                                                   const float* __restrict__ out_w,
                                                   const float* __restrict__ out_b,
                                                   float* __restrict__ l1) {
  const int row = blockIdx.x;
  const int tid = threadIdx.x;
  __shared__ int   sfeat[2][NN_NNZ];
  __shared__ float sval[NN_NNZ];
  __shared__ float swsum[8];

  if (tid < NN_NNZ) {
    sfeat[0][tid] = stm_idx[(row * NN_NNZ + tid) * 2 + 1];
  } else if (tid < 2 * NN_NNZ) {
    int k = tid - NN_NNZ;
    sfeat[1][k] = nstm_idx[(row * NN_NNZ + k) * 2 + 1];
  } else if (tid < 3 * NN_NNZ) {
    int k = tid - 2 * NN_NNZ;
    sval[k] = values[row * NN_NNZ + k];
  }
  __syncthreads();

  const int c = tid * 4; // this thread's 4 output channels
  float4 as = make_float4(0.f, 0.f, 0.f, 0.f);
  float4 an = make_float4(0.f, 0.f, 0.f, 0.f);
#pragma unroll 4
  for (int k = 0; k < NN_NNZ; ++k) {
    const float v = sval[k];
    const float4 ws = *(const float4*)(ftT + (size_t)sfeat[0][k] * NN_FTOUT + c);
    as.x = fmaf(v, ws.x, as.x); as.y = fmaf(v, ws.y, as.y);
    as.z = fmaf(v, ws.z, as.z); as.w = fmaf(v, ws.w, as.w);
    const float4 wn = *(const float4*)(ftT + (size_t)sfeat[1][k] * NN_FTOUT + c);
    an.x = fmaf(v, wn.x, an.x); an.y = fmaf(v, wn.y, an.y);
    an.z = fmaf(v, wn.z, an.z); an.w = fmaf(v, wn.w, an.w);
  }

  const float4 bb = *(const float4*)(ft_b + c);
  const float4 fb = *(const float4*)(fft_b + c);
  const float4 fs = *(const float4*)(fftC + (size_t)row * NN_FTOUT + c);
  const float4 fn = *(const float4*)(fftC + ((size_t)NN_B + row) * NN_FTOUT + c);
  const float4 ows = *(const float4*)(out_w + c);
  const float4 own = *(const float4*)(out_w + NN_FTOUT + c);

  float dot = 0.0f, h;
  h = fminf(fmaxf(as.x + bb.x + fb.x + fs.x, 0.f), 1.f); dot = fmaf(h, ows.x, dot);
  h = fminf(fmaxf(as.y + bb.y + fb.y + fs.y, 0.f), 1.f); dot = fmaf(h, ows.y, dot);
  h = fminf(fmaxf(as.z + bb.z + fb.z + fs.z, 0.f), 1.f); dot = fmaf(h, ows.z, dot);
  h = fminf(fmaxf(as.w + bb.w + fb.w + fs.w, 0.f), 1.f); dot = fmaf(h, ows.w, dot);
  h = fminf(fmaxf(an.x + bb.x + fb.x + fn.x, 0.f), 1.f); dot = fmaf(h, own.x, dot);
  h = fminf(fmaxf(an.y + bb.y + fb.y + fn.y, 0.f), 1.f); dot = fmaf(h, own.y, dot);
  h = fminf(fmaxf(an.z + bb.z + fb.z + fn.z, 0.f), 1.f); dot = fmaf(h, own.z, dot);
  h = fminf(fmaxf(an.w + bb.w + fb.w + fn.w, 0.f), 1.f); dot = fmaf(h, own.w, dot);

  // wave32 reduction, then cross-wave via LDS
#pragma unroll
  for (int off = 16; off > 0; off >>= 1) dot += __shfl_xor(dot, off, 32);
  if ((tid & 31) == 0) swsum[tid >> 5] = dot;
  __syncthreads();
  if (tid == 0) {
    float t = out_b[0];
#pragma unroll
    for (int w = 0; w < 8; ++w) t += swsum[w];
    l1[row] = t;
  }
}

// -------- bucket gather + sigmoid ------------------------------------------
__global__ __launch_bounds__(256) void nn_sigmoid(const float* __restrict__ l1,
                                                  const int* __restrict__ buckets,
                                                  float* __restrict__ out) {
  const int i = blockIdx.x * 256 + threadIdx.x;
  if (i < NN_B) {
    const int idx = buckets[i] + i; // BUCKET_COUNT == 1
    out[i] = 1.0f / (1.0f + expf(-l1[idx]));
  }
}

// ---------------------------------------------------------------------------
extern "C" void kernel_launch(void* const* d_in, const int* in_sizes, int n_in,
                              void* d_out, int out_size, void* d_ws, size_t ws_size,
                              hipStream_t stream) {
  (void)in_sizes; (void)n_in; (void)out_size;
  const int*   stm     = (const int*)d_in[0];
  const int*   nstm    = (const int*)d_in[1];
  const float* values  = (const float*)d_in[2];
  const int*   buckets = (const int*)d_in[3];
  const float* ft_w    = (const float*)d_in[4];
  const float* ft_b    = (const float*)d_in[5];
  const float* fft_w   = (const float*)d_in[6];
  const float* fft_b   = (const float*)d_in[7];
  const float* out_w   = (const float*)d_in[8];
  const float* out_b   = (const float*)d_in[9];

  float* ws = (float*)d_ws;
  size_t off = 0;
  float* ftT  = ws + off; off += (size_t)NN_FTIN * NN_FTOUT;   // 192 MB
  float* fftP = ws + off; off += (size_t)NN_VIRT * NN_FTOUT;   //   3 MB
  float* fftC = ws + off; off += 2ull * NN_B * NN_FTOUT;       //  32 MB
  float* l1   = ws + off; off += NN_B;
  if (ws_size < off * sizeof(float)) return; // workspace too small

  dim3 tb(32, 8);
  nn_transpose<<<dim3(NN_FTIN / 32, NN_FTOUT / 32), tb, 0, stream>>>(
      ft_w, ftT, NN_FTOUT, NN_FTIN);
  nn_pack_fft<<<(NN_VIRT * NN_FTOUT) / 256, 256, 0, stream>>>(fft_w, fftP);
  nn_fft_wmma<<<NN_B / 16, 256, 2 * 16 * VB_STRIDE * sizeof(float), stream>>>(
      stm, nstm, values, fftP, fftC);
  nn_ft_final<<<NN_B, 256, 0, stream>>>(stm, nstm, values, ftT, ft_b, fft_b,
                                        fftC, out_w, out_b, l1);
  nn_sigmoid<<<NN_B / 256, 256, 0, stream>>>(l1, buckets, (float*)d_out);
}